// VectorQuantizer_60748017435021
// MI455X (gfx1250) — compile-verified
//
#include <hip/hip_runtime.h>
#include <hip/hip_bf16.h>

typedef __attribute__((ext_vector_type(2))) float v2f;
typedef __attribute__((ext_vector_type(4))) float v4f;
typedef __attribute__((ext_vector_type(8))) float v8f;

#define K_EMB   4096
#define C_DIM   256
#define L_PER_B 4096
#define L_TOT   16384     // 4 * 64 * 64

#define BM 128            // rows (pixels) per workgroup
#define BN 64             // codebook cols per workgroup
#define CK 64             // K-chunk (channels) staged in LDS
#define LDSTR 68          // padded LDS row stride in floats (64 data + 4 pad)
#define NTHREADS 128      // 4 wave32; wave w owns rows [32w, 32w+32)

// monotonic float -> uint mapping (total order matching float <)
__device__ __forceinline__ unsigned int ordf(float f) {
  unsigned int u = __float_as_uint(f);
  return (u & 0x80000000u) ? ~u : (u | 0x80000000u);
}

__device__ __forceinline__ unsigned long long shflx_u64(unsigned long long v, int m) {
  unsigned int lo = (unsigned int)v;
  unsigned int hi = (unsigned int)(v >> 32);
  lo = (unsigned int)__shfl_xor((int)lo, m, 32);
  hi = (unsigned int)__shfl_xor((int)hi, m, 32);
  return ((unsigned long long)hi << 32) | (unsigned long long)lo;
}

// Pass 0: ||x||^2 per pixel, ||e||^2 per code, init argmin keys.
__global__ void vq_prep(const float* __restrict__ x, const float* __restrict__ emb,
                        float* __restrict__ xsq, float* __restrict__ esq,
                        unsigned long long* __restrict__ amin) {
  int tid = blockIdx.x * blockDim.x + threadIdx.x;
  if (tid < L_TOT) {
    int b = tid >> 12;
    int l = tid & (L_PER_B - 1);
    const float* p = x + (size_t)b * (C_DIM * L_PER_B) + l;
    float s = 0.f;
#pragma unroll 4
    for (int c = 0; c < C_DIM; ++c) {
      float v = p[(size_t)c * L_PER_B];
      s += v * v;
    }
    xsq[tid] = s;
  } else if (tid < L_TOT + K_EMB) {
    int k = tid - L_TOT;
    const v4f* p = (const v4f*)(emb + (size_t)k * C_DIM);
    float s = 0.f;
#pragma unroll 4
    for (int c = 0; c < C_DIM / 4; ++c) {
      v4f v = p[c];
      s += v.x * v.x + v.y * v.y + v.z * v.z + v.w * v.w;
    }
    esq[k] = s;
  } else if (tid < L_TOT + K_EMB + L_TOT) {
    amin[tid - (L_TOT + K_EMB)] = ~0ull;
  }
}

// Pass 1: fp32 WMMA GEMM -> distances + fused argmin via atomic-min on ordered keys.
// 4 waves; each wave register-blocks 2 M-tiles x 4 N-tiles (64 acc VGPRs):
// inner step = 6 ds_load_b64 : 8 v_wmma (vs 5:4 with 1 M-tile).
__global__ void __launch_bounds__(NTHREADS)
vq_gemm(const float* __restrict__ x, const float* __restrict__ emb,
        const float* __restrict__ xsq, const float* __restrict__ esq,
        float* __restrict__ dist, unsigned long long* __restrict__ amin) {
  __shared__ float Xs[BM * LDSTR];  // [l_local][c_local], stride 68
  __shared__ float Es[BN * LDSTR];  // [k_local][c_local], stride 68

  const int t    = threadIdx.x;
  const int wave = t >> 5;          // 0..3
  const int lane = t & 31;
  const int n16  = lane & 15;       // m for A-frag, n for B-frag / D
  const int h    = lane >> 4;       // lane half

  const int mIdx = blockIdx.x & 127;   // 128 M-blocks (fast dim: reuse E tile in L2)
  const int nIdx = blockIdx.x >> 7;    // 64 N-blocks
  const int r0   = mIdx * BM;          // global flat row base (b*4096 + l)
  const int k0   = nIdx * BN;

  // All BM rows of this block live in one batch (128 | 4096).
  const float* xb = x + (size_t)(r0 >> 12) * (C_DIM * L_PER_B) + (r0 & (L_PER_B - 1));

  const v8f vzero = {0.f, 0.f, 0.f, 0.f, 0.f, 0.f, 0.f, 0.f};
  v8f acc[2][4] = {{vzero, vzero, vzero, vzero}, {vzero, vzero, vzero, vzero}};

  // cooperative-load thread mapping (128 threads)
  const int xl  = t;                  // row 0..127 of X tile (all 64 c per thread)
  const int ek  = t >> 1;             // row 0..63 of E tile
  const int ecp = (t & 1) * 32;       // c-part 0/32
  const float* ebase = emb + (size_t)(k0 + ek) * C_DIM + ecp;

  for (int c0 = 0; c0 < C_DIM; c0 += CK) {
    __syncthreads();
    // X tile: transpose [c][l]-major global -> [l][c] LDS (4x b32 load + b128 store)
#pragma unroll
    for (int i = 0; i < CK; i += 4) {
      float v0 = xb[(size_t)(c0 + i + 0) * L_PER_B + xl];
      float v1 = xb[(size_t)(c0 + i + 1) * L_PER_B + xl];
      float v2 = xb[(size_t)(c0 + i + 2) * L_PER_B + xl];
      float v3 = xb[(size_t)(c0 + i + 3) * L_PER_B + xl];
      v4f vv = {v0, v1, v2, v3};
      *(v4f*)&Xs[xl * LDSTR + i] = vv;
    }
    // E tile: already row-major along c -> b128 in / b128 out
#pragma unroll
    for (int i = 0; i < 32; i += 4) {
      *(v4f*)&Es[ek * LDSTR + ecp + i] = *(const v4f*)&ebase[c0 + i];
    }
    __syncthreads();

    // WMMA K-loop. fp32 16x16x4 layout: A(m,k)/B(k,n) with k = vgpr + 2*laneHalf.
    const float* xrow0 = &Xs[(wave * 32 + n16) * LDSTR];       // M-tile 0
    const float* xrow1 = &Xs[(wave * 32 + 16 + n16) * LDSTR];  // M-tile 1
#pragma unroll 4
    for (int kk = 0; kk < CK; kk += 4) {
      v2f a0 = *(const v2f*)&xrow0[kk + 2 * h];
      v2f a1 = *(const v2f*)&xrow1[kk + 2 * h];
#pragma unroll
      for (int nt = 0; nt < 4; ++nt) {
        v2f bf = *(const v2f*)&Es[(nt * 16 + n16) * LDSTR + kk + 2 * h];
        acc[0][nt] = __builtin_amdgcn_wmma_f32_16x16x4_f32(
            false, a0, false, bf, (short)0, acc[0][nt], false, false);
        acc[1][nt] = __builtin_amdgcn_wmma_f32_16x16x4_f32(
            false, a1, false, bf, (short)0, acc[1][nt], false, false);
      }
    }
  }

  // Epilogue: dist = ||x||^2 + ||e||^2 - 2*dot ; store + fused argmin.
  float es[4];
#pragma unroll
  for (int nt = 0; nt < 4; ++nt) es[nt] = esq[k0 + nt * 16 + n16];

#pragma unroll
  for (int mt = 0; mt < 2; ++mt) {
#pragma unroll
    for (int i = 0; i < 8; ++i) {
      // D layout: vgpr i, laneHalf h -> M = i + 8h ; N = lane&15
      const int r = r0 + wave * 32 + mt * 16 + i + 8 * h;
      const float xs = xsq[r];
      unsigned long long best = ~0ull;
#pragma unroll
      for (int nt = 0; nt < 4; ++nt) {
        const int kcol = k0 + nt * 16 + n16;
        float d = xs + es[nt] - 2.0f * acc[mt][nt][i];
        dist[(size_t)r * K_EMB + kcol] = d;
        unsigned long long key =
            ((unsigned long long)ordf(d) << 32) | (unsigned long long)(unsigned int)kcol;
        best = best < key ? best : key;
      }
      // reduce over the 16 lanes of the N dimension (stays within each lane-half)
#pragma unroll
      for (int s = 1; s < 16; s <<= 1) {
        unsigned long long o = shflx_u64(best, s);
        best = best < o ? best : o;
      }
      if (n16 == 0) atomicMin(&amin[r], best);
    }
  }
}

// Pass 2: unpack argmin keys -> float indices (first L_TOT elements of d_out).
__global__ void vq_finish(const unsigned long long* __restrict__ amin,
                          float* __restrict__ outIdx) {
  int r = blockIdx.x * blockDim.x + threadIdx.x;
  if (r < L_TOT) outIdx[r] = (float)(unsigned int)(amin[r] & 0xffffffffull);
}

extern "C" void kernel_launch(void* const* d_in, const int* in_sizes, int n_in,
                              void* d_out, int out_size, void* d_ws, size_t ws_size,
                              hipStream_t stream) {
  (void)in_sizes; (void)n_in; (void)out_size; (void)ws_size;
  const float* x   = (const float*)d_in[0];   // [4, 256, 64, 64]
  const float* emb = (const float*)d_in[1];   // [4096, 256]
  float* out  = (float*)d_out;                // [16384 indices][16384*4096 distances]
  float* dist = out + L_TOT;

  // workspace: xsq f32[16384] | esq f32[4096] | argmin u64[16384]  (~208 KB)
  float* xsq = (float*)d_ws;
  float* esq = xsq + L_TOT;
  unsigned long long* amin = (unsigned long long*)(esq + K_EMB);

  vq_prep<<<(L_TOT + K_EMB + L_TOT + 255) / 256, 256, 0, stream>>>(x, emb, xsq, esq, amin);
  vq_gemm<<<(L_TOT / BM) * (K_EMB / BN), NTHREADS, 0, stream>>>(x, emb, xsq, esq, dist, amin);
  vq_finish<<<(L_TOT + 255) / 256, 256, 0, stream>>>(amin, out);
}